// SOCA_90701119357020
// MI455X (gfx1250) — compile-verified
//
#include <hip/hip_runtime.h>
#include <math.h>

typedef float v2f __attribute__((ext_vector_type(2)));
typedef float v8f __attribute__((ext_vector_type(8)));

constexpr int BATCH = 32;
constexpr int CH    = 256;     // channels
constexpr int SPAT  = 4096;    // 64*64
constexpr int RED   = 8;       // bottleneck reduction

// Async global->LDS copy, GVS mode: mem = SGPR64 + VGPR32 + inst_offset,
// LDS  = VGPR(lds) + inst_offset. Tracked by ASYNCcnt.
#define ASYNC_B128(ldsoff, voff, sbase, OFF)                                   \
  asm volatile("global_load_async_to_lds_b128 %0, %1, %2 offset:" OFF          \
               :: "v"(ldsoff), "v"(voff), "s"(sbase) : "memory")
#define WAIT_ASYNC(N) asm volatile("s_wait_asynccnt " N ::: "memory")

// ---------------------------------------------------------------------------
// 1) per-(b,c) spatial mean
// ---------------------------------------------------------------------------
__global__ void mean_kernel(const float* __restrict__ x, float* __restrict__ mean) {
  __shared__ float red[256];
  const int bc = blockIdx.x;                      // 0..BATCH*CH-1
  const float* p = x + (size_t)bc * SPAT;
  float s = 0.f;
  for (int i = threadIdx.x; i < SPAT; i += 256) s += p[i];
  red[threadIdx.x] = s;
  __syncthreads();
  for (int off = 128; off > 0; off >>= 1) {
    if (threadIdx.x < off) red[threadIdx.x] += red[threadIdx.x + off];
    __syncthreads();
  }
  if (threadIdx.x == 0) mean[bc] = red[0] * (1.0f / SPAT);
}

// ---------------------------------------------------------------------------
// 2) cov[b] = (1/M) * x * x^T - mu mu^T   (centered via rank-1 correction)
//    Raw x panels streamed global->LDS with async B128 copies, double
//    buffered so WMMA of chunk k overlaps the async fill of chunk k+1.
//    grid: (16 tiles of 64x64, BATCH)
// ---------------------------------------------------------------------------
__global__ void cov_kernel(const float* __restrict__ x, const float* __restrict__ mean,
                           float* __restrict__ cov) {
  __shared__ float As2[2][64][68];   // As2[buf][r][k] = x[i0+r][k0+k]
  __shared__ float Bs2[2][64][68];   // Bs2[buf][r][k] = x[j0+r][k0+k]
  const int b    = blockIdx.y;
  const int tile = blockIdx.x;
  const int i0   = (tile >> 2) * 64;
  const int j0   = (tile & 3) * 64;
  const int tid  = threadIdx.x;
  const int lane = tid & 31;
  const int wave = tid >> 5;
  const int wr   = (wave >> 2) * 32;
  const int wc   = (wave & 3) * 16;

  v8f acc0 = {}, acc1 = {};

  // panel-fill assignment: each thread owns 16 consecutive floats of one row
  const int rowL = tid >> 2;            // 0..63
  const int kq   = (tid & 3) * 16;

  const unsigned long long sA =
      (unsigned long long)(const void*)(x + (size_t)(b * CH + i0) * SPAT);
  const unsigned long long sB =
      (unsigned long long)(const void*)(x + (size_t)(b * CH + j0) * SPAT);
  const unsigned aLds0 = (unsigned)(uintptr_t)&As2[0][rowL][kq];
  const unsigned aLds1 = (unsigned)(uintptr_t)&As2[1][rowL][kq];
  const unsigned bLds0 = (unsigned)(uintptr_t)&Bs2[0][rowL][kq];
  const unsigned bLds1 = (unsigned)(uintptr_t)&Bs2[1][rowL][kq];

  // fragment addressing
  const int arow = wr + (lane & 15);
  const int koff = (lane >> 4) * 2;     // lanes 16-31 hold K=2,3 of each step
  const int bcol = wc + (lane & 15);

  constexpr int NCHUNK = SPAT / 64;

  // prologue: fill buffer 0 with chunk 0 (8 async B128 per thread-group step)
  {
    const unsigned vo = (unsigned)((rowL * SPAT + kq) * 4);
    ASYNC_B128(aLds0, vo, sA, "0");  ASYNC_B128(aLds0, vo, sA, "16");
    ASYNC_B128(aLds0, vo, sA, "32"); ASYNC_B128(aLds0, vo, sA, "48");
    ASYNC_B128(bLds0, vo, sB, "0");  ASYNC_B128(bLds0, vo, sB, "16");
    ASYNC_B128(bLds0, vo, sB, "32"); ASYNC_B128(bLds0, vo, sB, "48");
  }

  for (int c = 0; c < NCHUNK; ++c) {
    const int cur = c & 1;
    if (c + 1 < NCHUNK) {
      const unsigned vo = (unsigned)((rowL * SPAT + (c + 1) * 64 + kq) * 4);
      const unsigned al = cur ? aLds0 : aLds1;
      const unsigned bl = cur ? bLds0 : bLds1;
      ASYNC_B128(al, vo, sA, "0");  ASYNC_B128(al, vo, sA, "16");
      ASYNC_B128(al, vo, sA, "32"); ASYNC_B128(al, vo, sA, "48");
      ASYNC_B128(bl, vo, sB, "0");  ASYNC_B128(bl, vo, sB, "16");
      ASYNC_B128(bl, vo, sB, "32"); ASYNC_B128(bl, vo, sB, "48");
      WAIT_ASYNC("0x8");            // in-order: current chunk's 8 are done
    } else {
      WAIT_ASYNC("0x0");
    }
    __syncthreads();                // all waves' async fills of `cur` visible

    const float (*Ap)[68] = As2[cur];
    const float (*Bp)[68] = Bs2[cur];
#pragma unroll
    for (int kk = 0; kk < 64; kk += 4) {
      v2f a0, a1, bb;
      a0.x = Ap[arow][kk + koff];
      a0.y = Ap[arow][kk + koff + 1];
      a1.x = Ap[arow + 16][kk + koff];
      a1.y = Ap[arow + 16][kk + koff + 1];
      bb.x = Bp[bcol][kk + koff];         // B[k][n] = x[j0+n][k]
      bb.y = Bp[bcol][kk + koff + 1];
      acc0 = __builtin_amdgcn_wmma_f32_16x16x4_f32(false, a0, false, bb,
                                                   (short)0, acc0, false, false);
      acc1 = __builtin_amdgcn_wmma_f32_16x16x4_f32(false, a1, false, bb,
                                                   (short)0, acc1, false, false);
    }
    __syncthreads();                // done reading before `cur` is refilled
  }

  // epilogue: scale + rank-1 mean correction
  const float muJ  = mean[b * CH + j0 + wc + (lane & 15)];
  const int   col  = j0 + wc + (lane & 15);
  const int  rbase = i0 + wr + 8 * (lane >> 4);
#pragma unroll
  for (int v = 0; v < 8; ++v) {
    const float mi0 = mean[b * CH + rbase + v];
    const float mi1 = mean[b * CH + rbase + 16 + v];
    cov[((size_t)b * CH + rbase + v) * CH + col]      = acc0[v] * (1.0f / SPAT) - mi0 * muJ;
    cov[((size_t)b * CH + rbase + 16 + v) * CH + col] = acc1[v] * (1.0f / SPAT) - mi1 * muJ;
  }
}

// ---------------------------------------------------------------------------
// 3) normA[b] = sum of all entries of cov[b]
// ---------------------------------------------------------------------------
__global__ void norm_kernel(const float* __restrict__ cov, float* __restrict__ normA) {
  __shared__ float red[256];
  const int b = blockIdx.x;
  const float* p = cov + (size_t)b * CH * CH;
  float s = 0.f;
  for (int i = threadIdx.x; i < CH * CH; i += 256) s += p[i];
  red[threadIdx.x] = s;
  __syncthreads();
  for (int off = 128; off > 0; off >>= 1) {
    if (threadIdx.x < off) red[threadIdx.x] += red[threadIdx.x + off];
    __syncthreads();
  }
  if (threadIdx.x == 0) normA[b] = red[0];
}

// ---------------------------------------------------------------------------
// 4) A = cov / normA (in place), Z = 0.5*(3I - A)
// ---------------------------------------------------------------------------
__global__ void aZ_kernel(float* __restrict__ cov, const float* __restrict__ normA,
                          float* __restrict__ Z) {
  const size_t idx = (size_t)blockIdx.x * 256 + threadIdx.x;   // < BATCH*CH*CH
  const int b  = (int)(idx >> 16);
  const int ij = (int)(idx & 65535);
  const int i = ij >> 8, j = ij & 255;
  const float a = cov[idx] / normA[b];
  cov[idx] = a;
  Z[idx] = 0.5f * ((i == j ? 3.f : 0.f) - a);
}

// ---------------------------------------------------------------------------
// 5) batched D = scale * opA(A) @ opB(Bm); op = (3I - X) when t? != 0
//    grid: (16 tiles, BATCH); operands are L2-resident (0.8MB/batch)
// ---------------------------------------------------------------------------
__global__ void bmm_kernel(const float* __restrict__ A, const float* __restrict__ Bm,
                           float* __restrict__ D, int ta, int tb, float scale) {
  __shared__ float As[64][68];
  __shared__ float Bs[64][68];      // Bs[k][j]
  const int b    = blockIdx.y;
  const int tile = blockIdx.x;
  const int i0   = (tile >> 2) * 64;
  const int j0   = (tile & 3) * 64;
  const int tid  = threadIdx.x;
  const int lane = tid & 31;
  const int wave = tid >> 5;
  const int wr   = (wave >> 2) * 32;
  const int wc   = (wave & 3) * 16;

  v8f acc0 = {}, acc1 = {};

  const int rowL = tid >> 2;
  const int kq   = (tid & 3) * 16;

  const int arow = wr + (lane & 15);
  const int koff = (lane >> 4) * 2;
  const int bcol = wc + (lane & 15);

  for (int k0 = 0; k0 < CH; k0 += 64) {
    {
      const int gi = i0 + rowL;
      const float* src = A + ((size_t)b * CH + gi) * CH + k0 + kq;
#pragma unroll
      for (int q = 0; q < 4; ++q) {
        float4 v = *(const float4*)(src + q * 4);
        if (ta) {
          const int gk = k0 + kq + q * 4;
          v.x = ((gi == gk + 0) ? 3.f : 0.f) - v.x;
          v.y = ((gi == gk + 1) ? 3.f : 0.f) - v.y;
          v.z = ((gi == gk + 2) ? 3.f : 0.f) - v.z;
          v.w = ((gi == gk + 3) ? 3.f : 0.f) - v.w;
        }
        *(float4*)&As[rowL][kq + q * 4] = v;
      }
    }
    {
      const int gk = k0 + rowL;
      const float* src = Bm + ((size_t)b * CH + gk) * CH + j0 + kq;
#pragma unroll
      for (int q = 0; q < 4; ++q) {
        float4 v = *(const float4*)(src + q * 4);
        if (tb) {
          const int gj = j0 + kq + q * 4;
          v.x = ((gk == gj + 0) ? 3.f : 0.f) - v.x;
          v.y = ((gk == gj + 1) ? 3.f : 0.f) - v.y;
          v.z = ((gk == gj + 2) ? 3.f : 0.f) - v.z;
          v.w = ((gk == gj + 3) ? 3.f : 0.f) - v.w;
        }
        *(float4*)&Bs[rowL][kq + q * 4] = v;
      }
    }
    __syncthreads();
#pragma unroll
    for (int kk = 0; kk < 64; kk += 4) {
      v2f a0, a1, bb;
      a0.x = As[arow][kk + koff];
      a0.y = As[arow][kk + koff + 1];
      a1.x = As[arow + 16][kk + koff];
      a1.y = As[arow + 16][kk + koff + 1];
      bb.x = Bs[kk + koff][bcol];
      bb.y = Bs[kk + koff + 1][bcol];
      acc0 = __builtin_amdgcn_wmma_f32_16x16x4_f32(false, a0, false, bb,
                                                   (short)0, acc0, false, false);
      acc1 = __builtin_amdgcn_wmma_f32_16x16x4_f32(false, a1, false, bb,
                                                   (short)0, acc1, false, false);
    }
    __syncthreads();
  }

  const int col   = j0 + wc + (lane & 15);
  const int rbase = i0 + wr + 8 * (lane >> 4);
#pragma unroll
  for (int v = 0; v < 8; ++v) {
    D[((size_t)b * CH + rbase + v) * CH + col]      = acc0[v] * scale;
    D[((size_t)b * CH + rbase + 16 + v) * CH + col] = acc1[v] * scale;
  }
}

// ---------------------------------------------------------------------------
// 6) s[b][j] = sqrt(normA[b]) * mean_i ZYf[b][i][j]
// ---------------------------------------------------------------------------
__global__ void colmean_kernel(const float* __restrict__ ZYf, const float* __restrict__ normA,
                               float* __restrict__ svec) {
  const int b = blockIdx.x, j = threadIdx.x;
  const float* p = ZYf + (size_t)b * CH * CH + j;
  float s = 0.f;
  for (int i = 0; i < CH; ++i) s += p[(size_t)i * CH];
  svec[b * CH + j] = s * (1.0f / CH) * sqrtf(normA[b]);
}

// ---------------------------------------------------------------------------
// 7) gate = sigmoid(relu(s @ w1^T + b1) @ w2^T + b2)
// ---------------------------------------------------------------------------
__global__ void fc_kernel(const float* __restrict__ svec,
                          const float* __restrict__ w1, const float* __restrict__ b1,
                          const float* __restrict__ w2, const float* __restrict__ b2,
                          float* __restrict__ gate) {
  __shared__ float sS[CH];
  __shared__ float part[CH / RED][8];
  __shared__ float hdn[CH / RED];
  const int b = blockIdx.x, t = threadIdx.x;
  sS[t] = svec[b * CH + t];
  __syncthreads();
  const int h = t >> 3, pp = t & 7;          // 8 threads per hidden unit
  float acc = 0.f;
  for (int c = pp * 32; c < pp * 32 + 32; ++c) acc += sS[c] * w1[h * CH + c];
  part[h][pp] = acc;
  __syncthreads();
  if (pp == 0) {
    float s = b1[h];
#pragma unroll
    for (int q = 0; q < 8; ++q) s += part[h][q];
    hdn[h] = fmaxf(s, 0.f);
  }
  __syncthreads();
  float g = b2[t];
#pragma unroll
  for (int q = 0; q < CH / RED; ++q) g += hdn[q] * w2[t * (CH / RED) + q];
  gate[b * CH + t] = 1.f / (1.f + __expf(-g));
}

// ---------------------------------------------------------------------------
// 8) out = gate[b,c] * x  (float4 vectorized)
// ---------------------------------------------------------------------------
__global__ void scale_kernel(const float* __restrict__ x, const float* __restrict__ gate,
                             float* __restrict__ out) {
  const size_t idx = ((size_t)blockIdx.x * 256 + threadIdx.x) * 4;
  const int bc = (int)(idx >> 12);            // SPAT = 4096 per (b,c)
  __builtin_prefetch(x + idx + 4096, 0, 1);
  const float g = gate[bc];
  float4 v = *(const float4*)(x + idx);
  v.x *= g; v.y *= g; v.z *= g; v.w *= g;
  *(float4*)(out + idx) = v;
}

// ---------------------------------------------------------------------------
extern "C" void kernel_launch(void* const* d_in, const int* in_sizes, int n_in,
                              void* d_out, int out_size, void* d_ws, size_t ws_size,
                              hipStream_t stream) {
  const float* x  = (const float*)d_in[0];
  const float* w1 = (const float*)d_in[1];
  const float* b1 = (const float*)d_in[2];
  const float* w2 = (const float*)d_in[3];
  const float* b2 = (const float*)d_in[4];
  float* out = (float*)d_out;

  float* ws = (float*)d_ws;
  const size_t MAT = (size_t)BATCH * CH * CH;      // 2,097,152 floats per matrix set
  float* mean  = ws;                               // BATCH*CH
  float* normA = mean + BATCH * CH;                // BATCH (+pad)
  float* svec  = normA + 64;                       // BATCH*CH
  float* gate  = svec + BATCH * CH;                // BATCH*CH
  float* base  = gate + BATCH * CH;
  float* buf[5];
  for (int i = 0; i < 5; ++i) buf[i] = base + (size_t)i * MAT;

  const dim3 gTile(16, BATCH);

  mean_kernel<<<BATCH * CH, 256, 0, stream>>>(x, mean);
  cov_kernel<<<gTile, 256, 0, stream>>>(x, mean, buf[0]);
  norm_kernel<<<BATCH, 256, 0, stream>>>(buf[0], normA);
  aZ_kernel<<<(int)(MAT / 256), 256, 0, stream>>>(buf[0], normA, buf[2]);

  // Y = A @ ZY = 0.5 * A @ (3I - A)
  bmm_kernel<<<gTile, 256, 0, stream>>>(buf[0], buf[0], buf[1], 0, 1, 0.5f);

  float* Y = buf[1];
  float* Z = buf[2];
  float* f0 = buf[0];   // A no longer needed -> scratch
  float* f1 = buf[3];
  float* f2 = buf[4];

  for (int it = 0; it < 3; ++it) {
    float* T = f0;
    bmm_kernel<<<gTile, 256, 0, stream>>>(Z, Y, T, 1, 0, 0.5f);   // ZY = 0.5*(3I-Z)@Y
    bmm_kernel<<<gTile, 256, 0, stream>>>(Y, T, f1, 0, 0, 1.0f);  // Y' = Y@ZY
    bmm_kernel<<<gTile, 256, 0, stream>>>(T, Z, f2, 0, 0, 1.0f);  // Z' = ZY@Z
    float* oldY = Y; float* oldZ = Z;
    Y = f1; Z = f2;
    f0 = oldY; f1 = oldZ; f2 = T;
  }

  float* t1  = f0;
  float* ZYf = f1;
  bmm_kernel<<<gTile, 256, 0, stream>>>(Y, Z, t1, 0, 1, 0.5f);    // 0.5*Y@(3I-Z)
  bmm_kernel<<<gTile, 256, 0, stream>>>(t1, Y, ZYf, 0, 0, 1.0f);  // @ Y

  colmean_kernel<<<BATCH, 256, 0, stream>>>(ZYf, normA, svec);
  fc_kernel<<<BATCH, 256, 0, stream>>>(svec, w1, b1, w2, b2, gate);
  scale_kernel<<<(int)((size_t)BATCH * CH * SPAT / (256 * 4)), 256, 0, stream>>>(x, gate, out);
}